// ArcSoftmaxLoss_74363063763043
// MI455X (gfx1250) — compile-verified
//
#include <hip/hip_runtime.h>
#include <hip/hip_bf16.h>
#include <cmath>

// ---------------------------------------------------------------------------
// ArcFace softmax loss, MI455X (gfx1250).
// Memory-bound streaming logsumexp: 204.8 MB @ 23.3 TB/s ~= 8.8 us floor.
// Data path: CDNA5 async global->LDS pipeline (global_load_async_to_lds_b128,
// ASYNCcnt), 4-buffer ring, 3 x 8 KB chunks in flight per block (3 KB/wave).
// Compute in exp2 domain with raw v_exp_f32 (no libm range fixup).
// ---------------------------------------------------------------------------

#define BATCH       512
#define NCLS        100000
#define MARGIN_F    0.5f
#define SCALE_F     64.0f

// y = x * 64 * log2(e)  -> exp(64*x - m) == exp2(y - my)
#define KSCALE      92.33248261973106f   // 64 * log2(e)
#define LN2F        0.6931471805599453f

#define THREADS     256
#define CHUNK_FLOATS 2048                       // 8 KB chunk (2 x b128 / lane)
#define CHUNK_BYTES  (CHUNK_FLOATS * 4)         // 8192
#define NCHUNK       (NCLS / CHUNK_FLOATS)      // 48 full chunks = 98304 floats
#define TAIL4_START  (NCHUNK * CHUNK_FLOATS / 4)// 24576 (float4 index)
#define NCLS4        (NCLS / 4)                 // 25000
#define RING         4                          // LDS ring buffers (32 KB)
#define DEPTH        3                          // chunks in flight

// ---- raw hardware transcendentals (no range fixup; args <= 0 here) --------
__device__ __forceinline__ float exp2r(float x) { return __builtin_amdgcn_exp2f(x); }
__device__ __forceinline__ float log2r(float x) { return __builtin_amdgcn_logf(x); }

// ---- CDNA5 async global->LDS DMA (ASYNCcnt) -------------------------------
// GVS addressing: mem = SADDR(64b sgpr pair) + VADDR(32b vgpr); LDS dest byte
// address (relative to wave LDS base) comes from the VDST vgpr.
__device__ __forceinline__ void async_load_b128(unsigned ldsByteOff,
                                                unsigned glbByteOff,
                                                const float* base) {
    asm volatile("global_load_async_to_lds_b128 %0, %1, %2"
                 :: "v"(ldsByteOff), "v"(glbByteOff),
                    "s"((unsigned long long)(uintptr_t)base)
                 : "memory");
}

template <int N>
__device__ __forceinline__ void wait_asynccnt() {
    asm volatile("s_wait_asynccnt %0" :: "i"(N) : "memory");
}

__device__ __forceinline__ void wait_dscnt0() {
    asm volatile("s_wait_dscnt 0" ::: "memory");
}

// Issue one 8 KB chunk (2 async b128 per lane) into ring buffer (n & 3).
__device__ __forceinline__ void issue_chunk(int n, unsigned laneByte,
                                            const float* row) {
    const unsigned ldsOff = (unsigned)(n & (RING - 1)) * CHUNK_BYTES + laneByte;
    const unsigned gOff   = (unsigned)n * CHUNK_BYTES + laneByte;
    async_load_b128(ldsOff,         gOff,         row);
    async_load_b128(ldsOff + 4096u, gOff + 4096u, row);
}

// ---- online logsumexp accumulate (exp2 domain) ----------------------------
__device__ __forceinline__ void acc4(float4 v, float& m, float& s) {
    float y0 = v.x * KSCALE, y1 = v.y * KSCALE;
    float y2 = v.z * KSCALE, y3 = v.w * KSCALE;
    float vmax = fmaxf(fmaxf(y0, y1), fmaxf(y2, y3));
    if (vmax > m) {                 // rare: rescale running sum
        s *= exp2r(m - vmax);
        m = vmax;
    }
    s += (exp2r(y0 - m) + exp2r(y1 - m)) + (exp2r(y2 - m) + exp2r(y3 - m));
}

// ---------------------------------------------------------------------------
// Kernel 1: one workgroup per row. Streams 400 KB through a 32 KB LDS ring
// via async DMA, online (m,s) logsumexp, analytic margin correction on the
// label column, per-row partial -> d_ws.
// ---------------------------------------------------------------------------
__global__ __launch_bounds__(THREADS, 2)
void ArcSoftmaxLoss_rowlse_kernel(const float* __restrict__ costh,
                                  const int*   __restrict__ label,
                                  float*       __restrict__ partial) {
    // Single shared object => LDS offset 0 (matches asm VDST byte offsets).
    __shared__ __align__(16) unsigned char ldsraw[RING * CHUNK_BYTES + 64];
    float* stage = (float*)ldsraw;                       // ring stage
    float* redm  = (float*)(ldsraw + RING * CHUNK_BYTES);
    float* reds  = redm + 8;

    const int b   = blockIdx.x;
    const int tid = threadIdx.x;
    const float* row = costh + (size_t)b * NCLS;

    const unsigned laneByte = (unsigned)tid * 16u;       // b128 per lane

    float m = -INFINITY;
    float s = 0.0f;

    // Prologue: DEPTH chunks in flight (NCHUNK = 48 >= 3).
    issue_chunk(0, laneByte, row);
    issue_chunk(1, laneByte, row);
    issue_chunk(2, laneByte, row);

    for (int c = 0; c < NCHUNK; ++c) {
        // Async loads complete in order; 2 instrs per chunk.
        if (c + 2 < NCHUNK)      wait_asynccnt<4>();     // chunk c landed
        else if (c + 1 < NCHUNK) wait_asynccnt<2>();
        else                     wait_asynccnt<0>();

        const float* bp = stage + (unsigned)(c & (RING - 1)) * CHUNK_FLOATS
                                + tid * 4;
        float4 v0 = *(const float4*)bp;
        float4 v1 = *(const float4*)(bp + 1024);
        acc4(v0, m, s);
        acc4(v1, m, s);

        if (c + DEPTH < NCHUNK) {
            wait_dscnt0();       // LDS reads retired before DMA reuses buffer
            issue_chunk(c + DEPTH, laneByte, row);
        }
    }

    // Tail: 1696 floats = 424 float4, direct global loads.
    const float4* row4 = (const float4*)row;
    for (int i4 = TAIL4_START + tid; i4 < NCLS4; i4 += THREADS) {
        acc4(row4[i4], m, s);
    }

    // Wave32 combine of (m, s) pairs.
#pragma unroll
    for (int off = 16; off > 0; off >>= 1) {
        float om = __shfl_xor(m, off, 32);
        float os = __shfl_xor(s, off, 32);
        float M  = fmaxf(m, om);
        s = s * exp2r(m - M) + os * exp2r(om - M);
        m = M;
    }

    const int wave = tid >> 5, lane = tid & 31;
    if (lane == 0) { redm[wave] = m; reds[wave] = s; }
    __syncthreads();

    if (tid == 0) {
        float M = redm[0];
#pragma unroll
        for (int w = 1; w < 8; ++w) M = fmaxf(M, redm[w]);
        float S = 0.0f;
#pragma unroll
        for (int w = 0; w < 8; ++w) S += reds[w] * exp2r(redm[w] - M);

        // Margin correction: swap the label column's contribution.
        const int   lab = label[b];
        const float cy  = row[lab];
        const float cym = cosf(acosf(cy) + MARGIN_F);
        S += exp2r(cym * KSCALE - M) - exp2r(cy * KSCALE - M);

        const float lse = (M + log2r(S)) * LN2F;         // back to natural log
        partial[b] = lse - SCALE_F * cym;                // lse - logits[b,label]
    }
}

// ---------------------------------------------------------------------------
// Kernel 2: deterministic fixed-order mean of 512 partials -> scalar loss.
// ---------------------------------------------------------------------------
__global__ __launch_bounds__(256)
void ArcSoftmaxLoss_reduce_kernel(const float* __restrict__ partial,
                                  float*       __restrict__ out) {
    __shared__ float acc[8];
    const int tid = threadIdx.x;
    float v = partial[tid] + partial[tid + 256];
#pragma unroll
    for (int off = 16; off > 0; off >>= 1) v += __shfl_xor(v, off, 32);
    if ((tid & 31) == 0) acc[tid >> 5] = v;
    __syncthreads();
    if (tid == 0) {
        float t = 0.0f;
#pragma unroll
        for (int w = 0; w < 8; ++w) t += acc[w];
        out[0] = t / (float)BATCH;
    }
}

extern "C" void kernel_launch(void* const* d_in, const int* in_sizes, int n_in,
                              void* d_out, int out_size, void* d_ws, size_t ws_size,
                              hipStream_t stream) {
    const float* costh = (const float*)d_in[0];
    const int*   label = (const int*)d_in[1];
    float* partial = (float*)d_ws;      // 512 floats of scratch

    ArcSoftmaxLoss_rowlse_kernel<<<BATCH, THREADS, 0, stream>>>(costh, label, partial);
    ArcSoftmaxLoss_reduce_kernel<<<1, 256, 0, stream>>>(partial, (float*)d_out);
}